// TokenCrossModule_60679297958462
// MI455X (gfx1250) — compile-verified
//
#include <hip/hip_runtime.h>
#include <hip/hip_bf16.h>

typedef __attribute__((ext_vector_type(2))) float v2f;
typedef __attribute__((ext_vector_type(8))) float v8f;

constexpr int BN  = 16;            // batch
constexpr int SD  = 512;           // seq
constexpr int E   = 1024;          // embed
constexpr int GG  = 8;             // groups
constexpr int EG  = E / GG;        // 128
constexpr int DIN = SD * E / GG;   // 65536
constexpr int KB  = 64;            // K-block staged in LDS
constexpr int LDP = KB + 2;        // LDS row pad (bank-conflict avoidance)
constexpr int NP  = 32;            // N-panel per workgroup (2 tiles per wave)
constexpr int KSPLIT = 8;          // GEMM1 K-split across blocks
constexpr int KCH = DIN / KSPLIT;  // 8192 K per chunk

__device__ __forceinline__ float gelu_exact(float v) {
    return 0.5f * v * (1.0f + erff(v * 0.70710678118654752f));
}

__device__ __forceinline__ v8f wmma_f32x4(v2f a, v2f b, v8f c) {
    // V_WMMA_F32_16X16X4_F32 : D = A(16x4,f32) * B(4x16,f32) + C(16x16,f32)
    return __builtin_amdgcn_wmma_f32_16x16x4_f32(false, a, false, b, (short)0, c,
                                                 false, false);
}

// ---------------------------------------------------------------------------
// GEMM1 (K-split): hpart[kc, r, n] = sum_{d in chunk kc} xg[r,d] * W1[n,d]
//   r in [0,128) = b*8+g ; xg[r,d] = x[b, d>>7, g*128 + (d&127)]
// grid = (E/NP = 32, KSPLIT = 8) = 256 blocks; block = 256 (8 waves).
// Each wave: one 16-row M-tile x two 16-col N-tiles (independent acc chains).
// Double-buffered LDS; W1 streamed from HBM exactly once; x re-reads hit L2.
// ---------------------------------------------------------------------------
__global__ __launch_bounds__(256)
void gemm1_kernel(const float* __restrict__ x, const float* __restrict__ W1,
                  float* __restrict__ hpart) {
    __shared__ float sA[2][128][LDP];
    __shared__ float sB[2][NP][LDP];

    const int tid  = threadIdx.x;
    const int wave = tid >> 5;
    const int lane = tid & 31;
    const int half = lane >> 4;
    const int l15  = lane & 15;
    const int n0   = blockIdx.x * NP;
    const int kc   = blockIdx.y;
    const int kc0  = kc * KCH;         // multiple of 8192 (so kc0 & 127 == 0)

    // Per-thread cooperative-load slots: q fixed, rows rowb + 16*f.
    const int rowb = tid >> 4;         // 0..15
    const int qcol = (tid & 15) * 4;   // 0..60

    // Global source pointers (incremental from here on).
    const float* aptr[8];
#pragma unroll
    for (int f = 0; f < 8; ++f) {
        int row = rowb + 16 * f;
        int bi = row >> 3, gi = row & 7;
        aptr[f] = x + (size_t)(bi * SD + (kc0 >> 7)) * E + gi * EG + qcol;
    }
    const float* bptr[2];
#pragma unroll
    for (int f = 0; f < 2; ++f) {
        int n = rowb + 16 * f;
        bptr[f] = W1 + (size_t)(n0 + n) * DIN + kc0 + qcol;
    }

    float4 ra[8], rb[2];
    auto loadRegs = [&](int kb) {
#pragma unroll
        for (int f = 0; f < 8; ++f) ra[f] = *(const float4*)aptr[f];
#pragma unroll
        for (int f = 0; f < 2; ++f) {
            rb[f] = *(const float4*)bptr[f];
            __builtin_prefetch(bptr[f] + 2 * KB, 0, 1);  // next HBM panel block
        }
        const int da = (kb & 1) ? (E - KB) : KB;  // x-regroup: +64 / +960 floats
#pragma unroll
        for (int f = 0; f < 8; ++f) aptr[f] += da;
#pragma unroll
        for (int f = 0; f < 2; ++f) bptr[f] += KB;
    };
    auto storeLds = [&](int buf) {
#pragma unroll
        for (int f = 0; f < 8; ++f) {
            float* d = &sA[buf][rowb + 16 * f][qcol];
            d[0] = ra[f].x; d[1] = ra[f].y; d[2] = ra[f].z; d[3] = ra[f].w;
        }
#pragma unroll
        for (int f = 0; f < 2; ++f) {
            float* d = &sB[buf][rowb + 16 * f][qcol];
            d[0] = rb[f].x; d[1] = rb[f].y; d[2] = rb[f].z; d[3] = rb[f].w;
        }
    };

    v8f acc0 = {}, acc1 = {};
    constexpr int NKB = KCH / KB;      // 128 K-blocks per chunk

    loadRegs(0);
    storeLds(0);
    __syncthreads();

    for (int kb = 0; kb < NKB; ++kb) {
        const int cur = kb & 1;
        const bool more = (kb + 1 < NKB);
        if (more) loadRegs(kb + 1);    // global loads overlap WMMA below

#pragma unroll
        for (int kk = 0; kk < KB; kk += 4) {
            v2f a  = *(const v2f*)&sA[cur][wave * 16 + l15][kk + 2 * half];
            v2f b0 = *(const v2f*)&sB[cur][l15][kk + 2 * half];
            v2f b1 = *(const v2f*)&sB[cur][16 + l15][kk + 2 * half];
            acc0 = wmma_f32x4(a, b0, acc0);   // independent chains
            acc1 = wmma_f32x4(a, b1, acc1);
        }
        if (more) {
            storeLds(cur ^ 1);
            __syncthreads();
        }
    }

    // Store partials (bias/GELU applied in the reduction pass).
    const int col0 = n0 + l15;
#pragma unroll
    for (int j = 0; j < 8; ++j) {
        int row = wave * 16 + half * 8 + j;
        float* dst = hpart + ((size_t)kc * 128 + row) * E;
        dst[col0]      = acc0[j];
        dst[col0 + 16] = acc1[j];
    }
}

// ---------------------------------------------------------------------------
// Reduce K-split partials + bias + exact GELU, LayerNorm over e, row means.
// grid = 128 rows, block = 256.
// ---------------------------------------------------------------------------
__global__ __launch_bounds__(256)
void ln_kernel(const float* __restrict__ hpart, const float* __restrict__ b1,
               const float* __restrict__ ln_g, const float* __restrict__ ln_b,
               float* __restrict__ hln, float* __restrict__ mrow) {
    __shared__ float r1[256];
    __shared__ float r2[256];
    const int r = blockIdx.x;
    const int t = threadIdx.x;

    float v[4], s = 0.f, s2 = 0.f;
#pragma unroll
    for (int i = 0; i < 4; ++i) {
        const int e = t + 256 * i;
        float acc = b1[e];
#pragma unroll
        for (int p = 0; p < KSPLIT; ++p)
            acc += hpart[((size_t)p * 128 + r) * E + e];
        acc = gelu_exact(acc);
        v[i] = acc;
        s += acc;
        s2 += acc * acc;
    }
    r1[t] = s; r2[t] = s2;
    __syncthreads();
    for (int o = 128; o > 0; o >>= 1) {
        if (t < o) { r1[t] += r1[t + o]; r2[t] += r2[t + o]; }
        __syncthreads();
    }
    const float mu  = r1[0] * (1.0f / E);
    const float var = r2[0] * (1.0f / E) - mu * mu;
    const float inv = rsqrtf(var + 1e-5f);
    __syncthreads();  // before reusing r1

    float ms = 0.f;
#pragma unroll
    for (int i = 0; i < 4; ++i) {
        int e = t + 256 * i;
        float o = (v[i] - mu) * inv * ln_g[e] + ln_b[e];
        hln[r * E + e] = o;
        ms += o;
    }
    r1[t] = ms;
    __syncthreads();
    for (int o = 128; o > 0; o >>= 1) {
        if (t < o) r1[t] += r1[t + o];
        __syncthreads();
    }
    if (t == 0) mrow[r] = r1[0] * (1.0f / E);
}

// ---------------------------------------------------------------------------
// Squeeze-excite MLP over group dim (tiny): m[16,8] -> 4 -> 8 -> sigmoid scale
// ---------------------------------------------------------------------------
__global__ void se_kernel(const float* __restrict__ mrow, const float* __restrict__ se_w1,
                          const float* __restrict__ bn1_g, const float* __restrict__ bn1_b,
                          const float* __restrict__ bn1_m, const float* __restrict__ bn1_v,
                          const float* __restrict__ se_w2, const float* __restrict__ bn2_g,
                          const float* __restrict__ bn2_b, const float* __restrict__ bn2_m,
                          const float* __restrict__ bn2_v, float* __restrict__ scale) {
    __shared__ float sm[16][8];
    __shared__ float z1[16][4];
    const int t = threadIdx.x;
    if (t < 128) sm[t >> 3][t & 7] = mrow[t];
    __syncthreads();
    if (t < 64) {
        int b = t >> 2, i = t & 3;
        float z = 0.f;
#pragma unroll
        for (int j = 0; j < 8; ++j) z += se_w1[i * 8 + j] * sm[b][j];
        z = (z - bn1_m[i]) * rsqrtf(bn1_v[i] + 1e-5f) * bn1_g[i] + bn1_b[i];
        z1[b][i] = gelu_exact(z);
    }
    __syncthreads();
    if (t < 128) {
        int b = t >> 3, j = t & 7;
        float z = 0.f;
#pragma unroll
        for (int i = 0; i < 4; ++i) z += se_w2[j * 4 + i] * z1[b][i];
        z = (z - bn2_m[j]) * rsqrtf(bn2_v[j] + 1e-5f) * bn2_g[j] + bn2_b[j];
        scale[t] = 1.0f / (1.0f + expf(-z));
    }
}

// ---------------------------------------------------------------------------
// GEMM2: y[r, n] = sum_e (hln[r,e]*scale[r]) * W2[n,e] + b2[n], scattered to
// out[b, n>>7, g*128 + (n&127)].  N = 65536, K = 1024.
// grid = DIN/NP = 2048 panels; W2 streamed once; hln (512 KB) lives in L2.
// ---------------------------------------------------------------------------
__global__ __launch_bounds__(256)
void gemm2_kernel(const float* __restrict__ hln, const float* __restrict__ scale,
                  const float* __restrict__ W2, const float* __restrict__ b2,
                  float* __restrict__ out) {
    __shared__ float sA[2][128][LDP];
    __shared__ float sB[2][NP][LDP];
    __shared__ float sScale[128];

    const int tid  = threadIdx.x;
    const int wave = tid >> 5;
    const int lane = tid & 31;
    const int half = lane >> 4;
    const int l15  = lane & 15;
    const int n0   = blockIdx.x * NP;

    const int rowb = tid >> 4;
    const int qcol = (tid & 15) * 4;

    if (tid < 128) sScale[tid] = scale[tid];
    __syncthreads();  // sScale visible before first storeLds

    const float* aptr[8];
#pragma unroll
    for (int f = 0; f < 8; ++f)
        aptr[f] = hln + (size_t)(rowb + 16 * f) * E + qcol;
    const float* bptr[2];
#pragma unroll
    for (int f = 0; f < 2; ++f)
        bptr[f] = W2 + (size_t)(n0 + rowb + 16 * f) * E + qcol;

    float4 ra[8], rb[2];
    auto loadRegs = [&]() {
#pragma unroll
        for (int f = 0; f < 8; ++f) { ra[f] = *(const float4*)aptr[f]; aptr[f] += KB; }
#pragma unroll
        for (int f = 0; f < 2; ++f) {
            rb[f] = *(const float4*)bptr[f];
            __builtin_prefetch(bptr[f] + 2 * KB, 0, 1);
            bptr[f] += KB;
        }
    };
    auto storeLds = [&](int buf) {
#pragma unroll
        for (int f = 0; f < 8; ++f) {
            const float sc = sScale[rowb + 16 * f];
            float* d = &sA[buf][rowb + 16 * f][qcol];
            d[0] = ra[f].x * sc; d[1] = ra[f].y * sc;
            d[2] = ra[f].z * sc; d[3] = ra[f].w * sc;
        }
#pragma unroll
        for (int f = 0; f < 2; ++f) {
            float* d = &sB[buf][rowb + 16 * f][qcol];
            d[0] = rb[f].x; d[1] = rb[f].y; d[2] = rb[f].z; d[3] = rb[f].w;
        }
    };

    v8f acc0 = {}, acc1 = {};
    constexpr int NKB = E / KB;        // 16

    loadRegs();
    storeLds(0);
    __syncthreads();

    for (int kb = 0; kb < NKB; ++kb) {
        const int cur = kb & 1;
        const bool more = (kb + 1 < NKB);
        if (more) loadRegs();

#pragma unroll
        for (int kk = 0; kk < KB; kk += 4) {
            v2f a  = *(const v2f*)&sA[cur][wave * 16 + l15][kk + 2 * half];
            v2f b0 = *(const v2f*)&sB[cur][l15][kk + 2 * half];
            v2f b1 = *(const v2f*)&sB[cur][16 + l15][kk + 2 * half];
            acc0 = wmma_f32x4(a, b0, acc0);
            acc1 = wmma_f32x4(a, b1, acc1);
        }
        if (more) {
            storeLds(cur ^ 1);
            __syncthreads();
        }
    }

    // Epilogue: + b2, permuted scatter to [b,s,e]
    const int na = n0 + l15;
    const int nb = na + 16;
    const float ba = b2[na];
    const float bbv = b2[nb];
#pragma unroll
    for (int j = 0; j < 8; ++j) {
        int row = wave * 16 + half * 8 + j;
        int bi = row >> 3, gi = row & 7;
        size_t base = (size_t)bi * SD * E + (size_t)gi * EG;
        out[base + (size_t)(na >> 7) * E + (na & 127)] = acc0[j] + ba;
        out[base + (size_t)(nb >> 7) * E + (nb & 127)] = acc1[j] + bbv;
    }
}

extern "C" void kernel_launch(void* const* d_in, const int* in_sizes, int n_in,
                              void* d_out, int out_size, void* d_ws, size_t ws_size,
                              hipStream_t stream) {
    const float* x     = (const float*)d_in[0];
    const float* W1    = (const float*)d_in[1];
    const float* b1    = (const float*)d_in[2];
    const float* W2    = (const float*)d_in[3];
    const float* b2    = (const float*)d_in[4];
    const float* ln_g  = (const float*)d_in[5];
    const float* ln_b  = (const float*)d_in[6];
    const float* se_w1 = (const float*)d_in[7];
    const float* bn1_g = (const float*)d_in[8];
    const float* bn1_b = (const float*)d_in[9];
    const float* bn1_m = (const float*)d_in[10];
    const float* bn1_v = (const float*)d_in[11];
    const float* se_w2 = (const float*)d_in[12];
    const float* bn2_g = (const float*)d_in[13];
    const float* bn2_b = (const float*)d_in[14];
    const float* bn2_m = (const float*)d_in[15];
    const float* bn2_v = (const float*)d_in[16];
    float* out = (float*)d_out;

    float* ws    = (float*)d_ws;
    float* hpart = ws;                            // KSPLIT*128*1024 = 4 MB
    float* hln   = hpart + (size_t)KSPLIT * 128 * E;  // 128*1024
    float* mrow  = hln + 128 * E;                 // 128
    float* scl   = mrow + 128;                    // 128

    gemm1_kernel<<<dim3(E / NP, KSPLIT), 256, 0, stream>>>(x, W1, hpart);
    ln_kernel<<<128, 256, 0, stream>>>(hpart, b1, ln_g, ln_b, hln, mrow);
    se_kernel<<<1, 128, 0, stream>>>(mrow, se_w1, bn1_g, bn1_b, bn1_m, bn1_v,
                                     se_w2, bn2_g, bn2_b, bn2_m, bn2_v, scl);
    gemm2_kernel<<<DIN / NP, 256, 0, stream>>>(hln, scl, W2, b2, out);
}